// Compressor_79688823210249
// MI455X (gfx1250) — compile-verified
//
#include <hip/hip_runtime.h>
#include <cstdint>

// ---------------------------------------------------------------------------
// Feed-forward compressor for MI455X (gfx1250, wave32).
//
//   pass 0: memsetAsync flag, grid-stride max|x| reduction -> ws[0] (bypass)
//   pass 1: chunked scan. Each lane (wave32) owns one 4096-sample chunk of one
//           channel and re-runs a 49152-sample warmup (contraction factor
//           (1-alpha_release)^W ~ 3.6e-5 for release=0.1s) so all 4096 chunks
//           run in parallel. Per-lane streams are staged through LDS with
//           double-buffered global_load_async_to_lds_b128 (+ s_wait_asynccnt),
//           read back with ds_load_b128.
// ---------------------------------------------------------------------------

#define SAMPLE_RATE 48000.0f
#define NCH   16
#define CHUNK 4096          // payload samples per lane
#define WARM  49152         // warmup samples (multiple of SEG); ~10.2 release taus
#define SEG   16            // samples staged per lane per async round (4 x B128)
#define TPB   256           // 8 wave32 waves per block
#define WAVES (TPB / 32)
#define WSEGS (WARM / SEG)

// ---- CDNA5 async-to-LDS helpers -------------------------------------------

__device__ __forceinline__ uint32_t ldsOffset(const void* p) {
  // generic LDS pointers carry the LDS byte offset in the low 32 bits
  return (uint32_t)(uintptr_t)p;
}

__device__ __forceinline__ void asyncLoad16B(uint32_t ldsOff, const float* gptr) {
  uint64_t ga = (uint64_t)(uintptr_t)gptr;
  asm volatile("global_load_async_to_lds_b128 %0, %1, off"
               :: "v"(ldsOff), "v"(ga)
               : "memory");
}

// Issue one 16-sample segment (4 x b128) for this lane, clamped to [0, T-4].
__device__ __forceinline__ void issueSeg(const float* chan, int T, int tSeg,
                                         uint32_t ldsBase) {
#pragma unroll
  for (int j = 0; j < 4; ++j) {
    int tj = tSeg + j * 4;
    tj = tj < 0 ? 0 : tj;
    tj = tj > T - 4 ? T - 4 : tj;
    asyncLoad16B(ldsBase + (uint32_t)(j * 16), chan + tj);
  }
}

// ---- pass 0: max|x| for the silence-bypass --------------------------------

__global__ void maxabs_kernel(const float* __restrict__ x, int n,
                              unsigned int* __restrict__ out) {
  __shared__ float sm[TPB];
  float m = 0.0f;
  for (int i = blockIdx.x * blockDim.x + threadIdx.x; i < n;
       i += gridDim.x * blockDim.x)
    m = fmaxf(m, fabsf(x[i]));
  sm[threadIdx.x] = m;
  __syncthreads();
  for (int s = TPB / 2; s > 0; s >>= 1) {
    if (threadIdx.x < s) sm[threadIdx.x] = fmaxf(sm[threadIdx.x], sm[threadIdx.x + s]);
    __syncthreads();
  }
  // all values >= 0 so uint compare == float compare
  if (threadIdx.x == 0) atomicMax(out, __float_as_uint(sm[0]));
}

// ---- pass 1: chunked envelope scan + output -------------------------------

__global__ __launch_bounds__(TPB) void compressor_scan(
    const float* __restrict__ audio,
    const float* __restrict__ pThr, const float* __restrict__ pRatio,
    const float* __restrict__ pAtt, const float* __restrict__ pRel,
    const float* __restrict__ pMk,
    float* __restrict__ out, int T,
    const unsigned int* __restrict__ maxBits) {
  // [wave][buf][lane][5 x float4] : 80B lane stride (16B-aligned, 2-way bank conflicts)
  __shared__ __align__(16) float4 stage[WAVES][2][32][SEG / 4 + 1];

  const int b = blockIdx.y;
  const int chunkIdx = blockIdx.x * TPB + threadIdx.x;
  const int chunksPerChan = (T + CHUNK - 1) / CHUNK;
  if (chunkIdx >= chunksPerChan) return;

  const int w    = threadIdx.x >> 5;
  const int lane = threadIdx.x & 31;

  const float thr = *pThr;
  const float kr  = 1.0f - 1.0f / *pRatio;                       // (1 - 1/ratio)
  const float aA  = 1.0f - __expf(-1.0f / (*pAtt * SAMPLE_RATE));
  const float aR  = 1.0f - __expf(-1.0f / (*pRel * SAMPLE_RATE));
  const float mk  = *pMk;
  const bool bypass = (__uint_as_float(*maxBits) < 1e-6f);

  const float* chan  = audio + (size_t)b * (size_t)T;
  float*       ochan = out   + (size_t)b * (size_t)T;

  const int chunkStart = chunkIdx * CHUNK;
  const int gstart     = chunkStart - WARM;
  const int payloadLen = min(CHUNK, T - chunkStart);
  const int nseg       = WSEGS + ((payloadLen + SEG - 1) / SEG);

  const uint32_t ldsA = ldsOffset(&stage[w][0][lane][0]);
  const uint32_t ldsB = ldsOffset(&stage[w][1][lane][0]);

  // prologue: prime both buffers
  issueSeg(chan, T, gstart, ldsA);
  if (nseg > 1) issueSeg(chan, T, gstart + SEG, ldsB);

  const float DBK = 6.0205999132796239f;    // 20/log2(10): dB from log2
  const float EXK = 0.16609640474436813f;   // log2(10)/20: exp2 from dB

  float g = 0.0f;

  for (int i = 0; i < nseg; ++i) {
    // wait for segment i's 4 async loads (in-order completion)
    if (i + 1 < nseg) { asm volatile("s_wait_asynccnt 0x4" ::: "memory"); }
    else              { asm volatile("s_wait_asynccnt 0x0" ::: "memory"); }

    const float4* reg = (i & 1) ? (const float4*)&stage[w][1][lane][0]
                                : (const float4*)&stage[w][0][lane][0];
    float4 a0 = reg[0], a1 = reg[1], a2 = reg[2], a3 = reg[3];
    // make sure our ds_loads retired before the async engine refills this buffer
    asm volatile("s_wait_dscnt 0x0" ::: "memory");

    if (i + 2 < nseg)
      issueSeg(chan, T, gstart + (i + 2) * SEG, (i & 1) ? ldsB : ldsA);

    const int tSeg = gstart + i * SEG;
    union { float4 v[4]; float f[16]; } u;
    u.v[0] = a0; u.v[1] = a1; u.v[2] = a2; u.v[3] = a3;

    if (i < WSEGS) {
      // warmup: envelope only (t<=0 masked to gr=0 -> g stays exactly 0)
#pragma unroll
      for (int k = 0; k < SEG; ++k) {
        const int t = tSeg + k;
        const float x    = u.f[k];
        const float xa   = fabsf(x) + 1e-8f;
        const float db   = DBK * __log2f(xa);
        const float over = fmaxf(db - thr, 0.0f);
        const float gr   = (t <= 0) ? 0.0f : over * kr;
        const float a    = (gr > g) ? aA : aR;
        g = fmaf(a, gr - g, g);
      }
    } else {
      // payload: envelope + gain + output (buffered into float4 stores)
#pragma unroll
      for (int q = 0; q < 4; ++q) {
        union { float4 v; float f[4]; } ob;
#pragma unroll
        for (int r = 0; r < 4; ++r) {
          const int k = q * 4 + r;
          const int t = tSeg + k;
          const float x    = u.f[k];
          const float xa   = fabsf(x) + 1e-8f;
          const float db   = DBK * __log2f(xa);
          const float over = fmaxf(db - thr, 0.0f);
          const float gr   = (t <= 0) ? 0.0f : over * kr;
          const float a    = (gr > g) ? aA : aR;
          g = fmaf(a, gr - g, g);
          const float gl  = exp2f(EXK * (mk - g));   // native v_exp_f32
          const float sgn = (x > 0.0f) ? 1.0f : ((x < 0.0f) ? -1.0f : 0.0f);
          ob.f[r] = bypass ? x : sgn * xa * gl;
        }
        const int t0 = tSeg + q * 4;
        if (t0 + 3 < T) {
          *(float4*)(ochan + t0) = ob.v;
        } else {
#pragma unroll
          for (int r = 0; r < 4; ++r)
            if (t0 + r < T) ochan[t0 + r] = ob.f[r];
        }
      }
    }
  }
}

// ---- host entry -----------------------------------------------------------

extern "C" void kernel_launch(void* const* d_in, const int* in_sizes, int n_in,
                              void* d_out, int out_size, void* d_ws, size_t ws_size,
                              hipStream_t stream) {
  const float* audio = (const float*)d_in[0];
  const float* thr   = (const float*)d_in[1];
  const float* ratio = (const float*)d_in[2];
  const float* att   = (const float*)d_in[3];
  const float* rel   = (const float*)d_in[4];
  const float* mk    = (const float*)d_in[5];
  float* out = (float*)d_out;

  const int n = in_sizes[0];
  const int T = n / NCH;

  (void)hipMemsetAsync(d_ws, 0, sizeof(unsigned int), stream);
  maxabs_kernel<<<512, TPB, 0, stream>>>(audio, n, (unsigned int*)d_ws);

  const int chunksPerChan = (T + CHUNK - 1) / CHUNK;
  dim3 grid((chunksPerChan + TPB - 1) / TPB, NCH);
  compressor_scan<<<grid, TPB, 0, stream>>>(audio, thr, ratio, att, rel, mk,
                                            out, T, (const unsigned int*)d_ws);
}